// GCN_31447750542019
// MI455X (gfx1250) — compile-verified
//
#include <hip/hip_runtime.h>
#include <hip/hip_bf16.h>
#include <math.h>

typedef float v2f __attribute__((ext_vector_type(2)));
typedef float v8f __attribute__((ext_vector_type(8)));

#define F_IN 128

// ---------- helpers ----------
__device__ __forceinline__ void atomicAddF(float* addr, float v) {
  // lowers to global_atomic_add_f32 (no CAS loop)
  unsafeAtomicAdd(addr, v);
}

__global__ void fill_f32(float* __restrict__ p, float v, int n) {
  int i = blockIdx.x * blockDim.x + threadIdx.x;
  if (i < n) p[i] = v;
}

// ---------- degree / norm ----------
__global__ void deg_accum(const int* __restrict__ dst, float* __restrict__ deg, int E) {
  int e = blockIdx.x * blockDim.x + threadIdx.x;
  if (e < E) atomicAddF(&deg[dst[e]], 1.0f);
}

__global__ void deg_to_dinv(float* __restrict__ deg, int n) {
  int i = blockIdx.x * blockDim.x + threadIdx.x;
  if (i < n) {
    float d = deg[i];
    deg[i] = (d > 0.0f) ? rsqrtf(d) : 0.0f;
  }
}

// ---------- layer-1 GEMM: h1 = X @ W1  (f32 WMMA 16x16x4) ----------
__global__ __launch_bounds__(256) void gemm1_wmma(const float* __restrict__ X,
                                                  const float* __restrict__ W1,
                                                  float* __restrict__ h1, int n_tiles) {
  __shared__ float sW[F_IN * 16];           // 8 KB: whole W1
  for (int i = threadIdx.x; i < F_IN * 16; i += 256) sW[i] = W1[i];
  __syncthreads();

  int wave = threadIdx.x >> 5;
  int lane = threadIdx.x & 31;
  int tile = blockIdx.x * 8 + wave;
  if (tile >= n_tiles) return;

  int half = lane >> 4;                     // 0: K=0,1   1: K=2,3
  int col  = lane & 15;                     // A row / B,C column
  const float* arow = X + (size_t)(tile * 16 + col) * F_IN + 2 * half;

  v8f c = {};
  #pragma unroll 4
  for (int kb = 0; kb < F_IN; kb += 4) {
    v2f a = *(const v2f*)(arow + kb);       // A[m][kb+2h], A[m][kb+2h+1]
    v2f b;
    b.x = sW[(kb + 2 * half) * 16 + col];   // B rows K / K+1 per lane half
    b.y = sW[(kb + 2 * half + 1) * 16 + col];
    c = __builtin_amdgcn_wmma_f32_16x16x4_f32(false, a, false, b, (short)0, c, false, false);
  }

  float* out = h1 + (size_t)tile * 16 * 16;
  #pragma unroll
  for (int v = 0; v < 8; ++v)
    out[(v + 8 * half) * 16 + col] = c[v];
}

// ---------- layer-2 GEMM: z = Hbn @ W2 (W2 zero-padded 16x8 -> 16x16) ----------
__global__ __launch_bounds__(256) void gemm2_wmma(const float* __restrict__ H,
                                                  const float* __restrict__ W2,
                                                  float* __restrict__ Z,
                                                  int n_tiles, int ncls) {
  __shared__ float sW[16 * 16];
  {
    int k = threadIdx.x >> 4, cc = threadIdx.x & 15;
    if (threadIdx.x < 256) sW[threadIdx.x] = (cc < ncls) ? W2[k * ncls + cc] : 0.0f;
  }
  __syncthreads();

  int wave = threadIdx.x >> 5;
  int lane = threadIdx.x & 31;
  int tile = blockIdx.x * 8 + wave;
  if (tile >= n_tiles) return;

  int half = lane >> 4;
  int col  = lane & 15;
  const float* arow = H + (size_t)(tile * 16 + col) * 16 + 2 * half;

  v8f c = {};
  #pragma unroll
  for (int kb = 0; kb < 16; kb += 4) {
    v2f a = *(const v2f*)(arow + kb);
    v2f b;
    b.x = sW[(kb + 2 * half) * 16 + col];
    b.y = sW[(kb + 2 * half + 1) * 16 + col];
    c = __builtin_amdgcn_wmma_f32_16x16x4_f32(false, a, false, b, (short)0, c, false, false);
  }

  if (col < ncls) {
    #pragma unroll
    for (int v = 0; v < 8; ++v)
      Z[(size_t)(tile * 16 + v + 8 * half) * ncls + col] = c[v];
  }
}

// scalar tail for rows not covered by 16-row tiles (N%16, unused when N%16==0)
__global__ void gemm_tail(const float* __restrict__ X, const float* __restrict__ W,
                          float* __restrict__ out, int row0, int nrows, int K, int Kout) {
  int idx = blockIdx.x * blockDim.x + threadIdx.x;
  if (idx >= nrows * Kout) return;
  int r = row0 + idx / Kout, cc = idx % Kout;
  float acc = 0.0f;
  for (int k = 0; k < K; ++k) acc += X[(size_t)r * K + k] * W[k * Kout + cc];
  out[(size_t)r * Kout + cc] = acc;
}

// ---------- aggregation: out[dst] += h[src]*dinv[src]*dinv[dst], self-loop + bias in init ----------
__global__ void agg_init(const float* __restrict__ h, const float* __restrict__ dinv,
                         const float* __restrict__ bias, float* __restrict__ out,
                         int n, int lgF) {
  int idx = blockIdx.x * blockDim.x + threadIdx.x;
  if (idx >= (n << lgF)) return;
  int i = idx >> lgF, j = idx & ((1 << lgF) - 1);
  float di = dinv[i];
  out[idx] = h[idx] * di * di + bias[j];
}

__global__ void agg_edges(const int* __restrict__ src, const int* __restrict__ dst,
                          const float* __restrict__ h, const float* __restrict__ dinv,
                          float* __restrict__ out, int E, int lgF) {
  int idx = blockIdx.x * blockDim.x + threadIdx.x;
  int e = idx >> lgF;
  if (e >= E) return;
  int j = idx & ((1 << lgF) - 1);
  int s = src[e], d = dst[e];
  float w = dinv[s] * dinv[d];
  atomicAddF(&out[((size_t)d << lgF) + j], h[((size_t)s << lgF) + j] * w);
}

// ---------- BatchNorm (training stats) + ReLU ----------
__global__ __launch_bounds__(256) void bn_stats(const float* __restrict__ x,
                                                float* __restrict__ stats, int total) {
  // stats[0..15]=sum, stats[16..31]=sumsq (pre-zeroed). Stride is a multiple of 16,
  // so each thread touches exactly one channel: j = tid & 15.
  __shared__ float ssum[16], ssq[16];
  if (threadIdx.x < 16) { ssum[threadIdx.x] = 0.0f; ssq[threadIdx.x] = 0.0f; }
  __syncthreads();
  int T = gridDim.x * 256;
  float s = 0.0f, q = 0.0f;
  for (int i = blockIdx.x * 256 + threadIdx.x; i < total; i += T) {
    float v = x[i];
    s += v; q += v * v;
  }
  int j = threadIdx.x & 15;
  atomicAdd(&ssum[j], s);
  atomicAdd(&ssq[j], q);
  __syncthreads();
  if (threadIdx.x < 16) {
    atomicAddF(&stats[threadIdx.x], ssum[threadIdx.x]);
    atomicAddF(&stats[16 + threadIdx.x], ssq[threadIdx.x]);
  }
}

__global__ void bn_finalize(const float* __restrict__ gamma, const float* __restrict__ beta,
                            float* __restrict__ stats, int n) {
  int j = threadIdx.x;
  if (j < 16) {
    float inv_n = 1.0f / (float)n;
    float mean = stats[j] * inv_n;
    float var  = stats[16 + j] * inv_n - mean * mean;
    float sc   = gamma[j] * rsqrtf(var + 1e-5f);
    stats[32 + j] = sc;                 // scale
    stats[48 + j] = beta[j] - mean * sc;// shift
  }
}

__global__ void bn_apply_relu(const float* __restrict__ x, const float* __restrict__ stats,
                              float* __restrict__ y, int total) {
  int idx = blockIdx.x * blockDim.x + threadIdx.x;
  if (idx >= total) return;
  int j = idx & 15;
  float v = x[idx] * stats[32 + j] + stats[48 + j];
  y[idx] = v > 0.0f ? v : 0.0f;
}

// ---------- log_softmax over classes, in place ----------
__global__ void logsoftmax_rows(float* __restrict__ out, int n, int ncls) {
  int i = blockIdx.x * blockDim.x + threadIdx.x;
  if (i >= n) return;
  float v[16];
  float m = -INFINITY;
  for (int j = 0; j < ncls; ++j) { v[j] = out[(size_t)i * ncls + j]; m = fmaxf(m, v[j]); }
  float s = 0.0f;
  for (int j = 0; j < ncls; ++j) s += __expf(v[j] - m);
  float lse = m + __logf(s);
  for (int j = 0; j < ncls; ++j) out[(size_t)i * ncls + j] = v[j] - lse;
}

// ---------- launch ----------
extern "C" void kernel_launch(void* const* d_in, const int* in_sizes, int n_in,
                              void* d_out, int out_size, void* d_ws, size_t ws_size,
                              hipStream_t stream) {
  const float* x     = (const float*)d_in[0];
  const float* W1    = (const float*)d_in[1];
  const float* b1    = (const float*)d_in[2];
  const float* gamma = (const float*)d_in[3];
  const float* beta  = (const float*)d_in[4];
  const float* W2    = (const float*)d_in[5];
  const float* b2    = (const float*)d_in[6];
  const int*   ei    = (const int*)d_in[7];

  const int N   = in_sizes[0] / F_IN;   // 100000
  const int HID = in_sizes[2];          // 16
  const int NC  = in_sizes[6];          // 8
  const int E   = in_sizes[7] / 2;      // 3200000
  const int* srcp = ei;
  const int* dstp = ei + E;

  float* ws    = (float*)d_ws;
  float* dinv  = ws;                          // N
  float* h1    = dinv + N;                    // N*16
  float* agg1  = h1 + (size_t)N * 16;         // N*16
  float* stats = agg1 + (size_t)N * 16;       // 64
  float* hbn   = h1;                          // reuse (h1 dead after agg1)
  float* z     = agg1;                        // reuse (agg1 dead after bn_apply)
  float* out   = (float*)d_out;               // N*NC

  auto cdiv = [](long long a, int b) { return (int)((a + b - 1) / b); };

  // degrees -> dinv (in place)
  fill_f32<<<cdiv(N, 256), 256, 0, stream>>>(dinv, 1.0f, N);           // self loops
  deg_accum<<<cdiv(E, 256), 256, 0, stream>>>(dstp, dinv, E);
  deg_to_dinv<<<cdiv(N, 256), 256, 0, stream>>>(dinv, N);

  const int tiles = N / 16;
  const int rem   = N % 16;

  // layer 1: h1 = X @ W1
  gemm1_wmma<<<cdiv(tiles, 8), 256, 0, stream>>>(x, W1, h1, tiles);
  if (rem)
    gemm_tail<<<cdiv((long long)rem * HID, 256), 256, 0, stream>>>(x, W1, h1, tiles * 16, rem, F_IN, HID);

  // aggregate 1 (self loop + b1 in init, edges via L2 atomics)
  agg_init<<<cdiv((long long)N * 16, 256), 256, 0, stream>>>(h1, dinv, b1, agg1, N, 4);
  agg_edges<<<cdiv((long long)E * 16, 256), 256, 0, stream>>>(srcp, dstp, h1, dinv, agg1, E, 4);

  // BatchNorm + ReLU
  fill_f32<<<1, 64, 0, stream>>>(stats, 0.0f, 64);
  bn_stats<<<256, 256, 0, stream>>>(agg1, stats, N * 16);
  bn_finalize<<<1, 32, 0, stream>>>(gamma, beta, stats, N);
  bn_apply_relu<<<cdiv((long long)N * 16, 256), 256, 0, stream>>>(agg1, stats, hbn, N * 16);

  // layer 2: z = Hbn @ W2 (WMMA, W2 padded to 16 cols)
  gemm2_wmma<<<cdiv(tiles, 8), 256, 0, stream>>>(hbn, W2, z, tiles, NC);
  if (rem)
    gemm_tail<<<cdiv((long long)rem * NC, 256), 256, 0, stream>>>(hbn, W2, z, tiles * 16, rem, HID, NC);

  // aggregate 2 directly into d_out
  agg_init<<<cdiv((long long)N * 8, 256), 256, 0, stream>>>(z, dinv, b2, out, N, 3);
  agg_edges<<<cdiv((long long)E * 8, 256), 256, 0, stream>>>(srcp, dstp, z, dinv, out, E, 3);

  // log_softmax in place
  logsoftmax_rows<<<cdiv(N, 256), 256, 0, stream>>>(out, N, NC);
}